// BigBirdEncoder_58969900974412
// MI455X (gfx1250) — compile-verified
//
#include <hip/hip_runtime.h>
#include <hip/hip_bf16.h>
#include <cstddef>

typedef __attribute__((ext_vector_type(2))) float v2f;
typedef __attribute__((ext_vector_type(8))) float v8f;

constexpr int B_  = 2;
constexpr int S_  = 4096;
constexpr int D_  = 512;
constexpr int H_  = 8;
constexpr int NB_ = 64;
constexpr int HD_ = 64;
constexpr int M_  = 60;
constexpr float NEGV = -1000000000.0f;

#define LDA 68   // padded LDS row stride (floats); 68*4B rows stay 16B aligned, 68%64=4 breaks bank conflicts
#define LQ  68

// ---------------- WMMA helpers (fp32, 16x16x4) ----------------

__device__ __forceinline__ v8f wmma4(v2f a, v2f b, v8f c) {
  return __builtin_amdgcn_wmma_f32_16x16x4_f32(
      /*neg_a=*/false, a, /*neg_b=*/false, b,
      /*c_mod=*/(short)0, c, /*reuse_a=*/false, /*reuse_b=*/false);
}

// A fragment: A[m][k], rows row0..row0+15, k = kk..kk+3.
// Lanes 0-15: K=kk,kk+1 ; lanes 16-31: K=kk+2,kk+3 (ISA 32-bit A 16x4 layout).
__device__ __forceinline__ v2f load_a(const float* sm, int row0, int kk, int ld, int lane) {
  int mrow = row0 + (lane & 15);
  int kx = kk + ((lane >> 4) << 1);
  v2f a;
  a.x = sm[mrow * ld + kx];
  a.y = sm[mrow * ld + kx + 1];
  return a;
}

// B fragment: B[k][n] from row-major [k][n] storage.
__device__ __forceinline__ v2f load_b(const float* sm, int kk, int col0, int ld, int lane) {
  int n = col0 + (lane & 15);
  int kx = kk + ((lane >> 4) << 1);
  v2f b;
  b.x = sm[kx * ld + n];
  b.y = sm[(kx + 1) * ld + n];
  return b;
}

// B fragment with transpose: B[k][n] = Kmat[n][k] (Kmat stored [n][k] row-major).
__device__ __forceinline__ v2f load_bt(const float* sm, int kk, int col0, int ld, int lane) {
  int n = col0 + (lane & 15);
  int kx = kk + ((lane >> 4) << 1);
  v2f b;
  b.x = sm[n * ld + kx];
  b.y = sm[n * ld + kx + 1];
  return b;
}

// ---------------- 1) Q/K/V projection GEMM: (8192x512)x(512x512) ----------------
// Writes out in (B,H,S,HD) layout, optional scale (q gets HD^-0.5).

__global__ __launch_bounds__(128) void proj_kernel(const float* __restrict__ X,
                                                   const float* __restrict__ W,
                                                   float* __restrict__ out,
                                                   float scale) {
  __shared__ float Xs[64 * LDA];
  __shared__ float Ws[64 * LDA];
  const int tid = threadIdx.x;
  const int wave = tid >> 5, lane = tid & 31;
  const int row0 = blockIdx.x * 64;
  const int col0 = blockIdx.y * 64;

  v8f acc[4] = {};

  for (int k0 = 0; k0 < D_; k0 += 64) {
    __syncthreads();
#pragma unroll
    for (int i = 0; i < 8; i++) {
      int f = tid + i * 128;           // float4 index 0..1023
      int r = f >> 4, c4 = f & 15;
      *(float4*)&Xs[r * LDA + c4 * 4] =
          *(const float4*)(X + (size_t)(row0 + r) * D_ + k0 + c4 * 4);
      *(float4*)&Ws[r * LDA + c4 * 4] =
          *(const float4*)(W + (size_t)(k0 + r) * D_ + col0 + c4 * 4);
    }
    __syncthreads();
    const int arow = wave * 16;
#pragma unroll 4
    for (int kk = 0; kk < 64; kk += 4) {
      v2f a = load_a(Xs, arow, kk, LDA, lane);
#pragma unroll
      for (int ct = 0; ct < 4; ct++) {
        v2f b = load_b(Ws, kk, ct * 16, LDA, lane);
        acc[ct] = wmma4(a, b, acc[ct]);
      }
    }
  }

  const int nn = lane & 15;
  const int rofs = (lane >> 4) << 3;
#pragma unroll
  for (int ct = 0; ct < 4; ct++) {
    int col = col0 + ct * 16 + nn;
    int hh = col >> 6, hd = col & 63;
#pragma unroll
    for (int r = 0; r < 8; r++) {
      int row = row0 + wave * 16 + r + rofs;
      int bb = row >> 12;          // / 4096
      int ss = row & 4095;
      out[(((size_t)bb * H_ + hh) * S_ + ss) * HD_ + hd] = acc[ct][r] * scale;
    }
  }
}

// ---------------- 2) dense rows attention (blocks 0,1,62,63 vs all keys) ----------------

__global__ __launch_bounds__(128) void dense_attn(const float* __restrict__ q,
                                                  const float* __restrict__ k,
                                                  const float* __restrict__ v,
                                                  const float* __restrict__ mask,
                                                  float* __restrict__ ctx) {
  __shared__ float Qs[64 * LQ];
  __shared__ float Ks[64 * LQ];
  __shared__ float Vs[64 * 64];
  __shared__ float Ps[4][16 * LQ];
  __shared__ float kmask[64];
  __shared__ float alpha_s[4][16];
  __shared__ float rl[4][16];

  const int tid = threadIdx.x, wave = tid >> 5, lane = tid & 31;
  const int wg = blockIdx.x;              // B*H*4
  const int which = wg & 3;
  const int bh = wg >> 2;
  const int bb = bh >> 3, hh = bh & 7;
  const int qb = (which < 2) ? which : (NB_ - 4 + which);   // 0,1,62,63

  const float* qbase = q + (((size_t)bb * H_ + hh) * S_ + qb * 64) * HD_;
#pragma unroll
  for (int i = 0; i < 8; i++) {
    int f = tid + i * 128;
    int r = f >> 4, c4 = f & 15;
    *(float4*)&Qs[r * LQ + c4 * 4] = *(const float4*)(qbase + r * 64 + c4 * 4);
  }

  float mrun = -1e30f, lsum = 0.f;
  v8f acc[4] = {};
  const int nn = lane & 15;
  const int rofs = (lane >> 4) << 3;
  float* pw = Ps[wave];

  for (int kb = 0; kb < NB_; kb++) {
    __syncthreads();
    const float* kbase = k + (((size_t)bb * H_ + hh) * S_ + kb * 64) * HD_;
    const float* vbase = v + (((size_t)bb * H_ + hh) * S_ + kb * 64) * HD_;
#pragma unroll
    for (int i = 0; i < 8; i++) {
      int f = tid + i * 128;
      int r = f >> 4, c4 = f & 15;
      *(float4*)&Ks[r * LQ + c4 * 4] = *(const float4*)(kbase + r * 64 + c4 * 4);
      *(float4*)&Vs[r * 64 + c4 * 4] = *(const float4*)(vbase + r * 64 + c4 * 4);
    }
    if (tid < 64) kmask[tid] = mask[(size_t)bb * S_ + kb * 64 + tid];
    if (kb + 1 < NB_) {
      const float* nk = k + (((size_t)bb * H_ + hh) * S_ + (kb + 1) * 64) * HD_;
      __builtin_prefetch(nk + tid * 32, 0, 0);   // global_prefetch next K block
    }
    __syncthreads();

    // scores = Q (16 rows per wave) x K^T  -> 4 col tiles of 16x16
    v8f sf[4] = {};
#pragma unroll 4
    for (int kk = 0; kk < 64; kk += 4) {
      v2f a = load_a(Qs, wave * 16, kk, LQ, lane);
#pragma unroll
      for (int ct = 0; ct < 4; ct++) {
        v2f b = load_bt(Ks, kk, ct * 16, LQ, lane);
        sf[ct] = wmma4(a, b, sf[ct]);
      }
    }
    // stage scores (D-frag layout -> row-major LDS, wave-private)
#pragma unroll
    for (int ct = 0; ct < 4; ct++)
#pragma unroll
      for (int r = 0; r < 8; r++)
        pw[(r + rofs) * LQ + ct * 16 + nn] = sf[ct][r];
    __syncthreads();

    // online softmax: lane l (<16) owns row l of this wave
    if (lane < 16) {
      float* prow = &pw[lane * LQ];
      float rmax = -1e30f;
      for (int c = 0; c < 64; c++) {
        float sc = prow[c] + (1.0f - kmask[c]) * NEGV;
        prow[c] = sc;
        rmax = fmaxf(rmax, sc);
      }
      float nm = fmaxf(mrun, rmax);
      float al = __expf(mrun - nm);
      float psum = 0.f;
      for (int c = 0; c < 64; c++) {
        float p = __expf(prow[c] - nm);
        prow[c] = p;
        psum += p;
      }
      lsum = lsum * al + psum;
      mrun = nm;
      alpha_s[wave][lane] = al;
    }
    __syncthreads();

    // rescale accumulator rows by alpha
#pragma unroll
    for (int ct = 0; ct < 4; ct++)
#pragma unroll
      for (int r = 0; r < 8; r++)
        acc[ct][r] *= alpha_s[wave][r + rofs];

    // acc += P x V
#pragma unroll 4
    for (int kk = 0; kk < 64; kk += 4) {
      v2f a = load_a(pw, 0, kk, LQ, lane);
#pragma unroll
      for (int ht = 0; ht < 4; ht++) {
        v2f b = load_b(Vs, kk, ht * 16, 64, lane);
        acc[ht] = wmma4(a, b, acc[ht]);
      }
    }
  }

  if (lane < 16) rl[wave][lane] = 1.0f / lsum;
  __syncthreads();

  float* obase = ctx + (((size_t)bb * H_ + hh) * S_ + qb * 64) * HD_;
#pragma unroll
  for (int ct = 0; ct < 4; ct++)
#pragma unroll
    for (int r = 0; r < 8; r++)
      obase[(wave * 16 + r + rofs) * 64 + ct * 16 + nn] =
          acc[ct][r] * rl[wave][r + rofs];
}

// ---------------- 3) middle blocks attention (band + global + random) ----------------

__global__ __launch_bounds__(128) void mid_attn(const float* __restrict__ q,
                                                const float* __restrict__ k,
                                                const float* __restrict__ v,
                                                const float* __restrict__ mask,
                                                const int* __restrict__ rand_blocks,
                                                float* __restrict__ ctx) {
  __shared__ float Qs[64 * LQ];
  __shared__ float Ks[64 * LQ];
  __shared__ float Vs[64 * 64];
  __shared__ float Ps[4][16 * LQ];
  __shared__ float kmask[64];
  __shared__ float alpha_s[4][16];
  __shared__ float rl[4][16];
  __shared__ int kbs[8];

  const int tid = threadIdx.x, wave = tid >> 5, lane = tid & 31;
  const int wg = blockIdx.x;              // (b*H + h)*M + mi
  const int mi = wg % M_;
  const int bh = wg / M_;
  const int bb = bh >> 3, hh = bh & 7;
  const int qb = mi + 2;

  if (tid == 0) {
    kbs[0] = qb - 1; kbs[1] = qb; kbs[2] = qb + 1;    // band
    kbs[3] = 0;      kbs[4] = NB_ - 1;                // global
    kbs[5] = rand_blocks[mi * 3 + 0];
    kbs[6] = rand_blocks[mi * 3 + 1];
    kbs[7] = rand_blocks[mi * 3 + 2];
  }

  const float* qbase = q + (((size_t)bb * H_ + hh) * S_ + qb * 64) * HD_;
#pragma unroll
  for (int i = 0; i < 8; i++) {
    int f = tid + i * 128;
    int r = f >> 4, c4 = f & 15;
    *(float4*)&Qs[r * LQ + c4 * 4] = *(const float4*)(qbase + r * 64 + c4 * 4);
  }
  float qm = 1.0f;
  if (lane < 16) qm = mask[(size_t)bb * S_ + qb * 64 + wave * 16 + lane];

  float mrun = -1e30f, lsum = 0.f;
  v8f acc[4] = {};
  const int nn = lane & 15;
  const int rofs = (lane >> 4) << 3;
  float* pw = Ps[wave];

  for (int kbi = 0; kbi < 8; kbi++) {
    __syncthreads();
    const int kb = kbs[kbi];
    const float* kbase = k + (((size_t)bb * H_ + hh) * S_ + kb * 64) * HD_;
    const float* vbase = v + (((size_t)bb * H_ + hh) * S_ + kb * 64) * HD_;
#pragma unroll
    for (int i = 0; i < 8; i++) {
      int f = tid + i * 128;
      int r = f >> 4, c4 = f & 15;
      *(float4*)&Ks[r * LQ + c4 * 4] = *(const float4*)(kbase + r * 64 + c4 * 4);
      *(float4*)&Vs[r * 64 + c4 * 4] = *(const float4*)(vbase + r * 64 + c4 * 4);
    }
    if (tid < 64) kmask[tid] = mask[(size_t)bb * S_ + kb * 64 + tid];
    __syncthreads();

    v8f sf[4] = {};
#pragma unroll 4
    for (int kk = 0; kk < 64; kk += 4) {
      v2f a = load_a(Qs, wave * 16, kk, LQ, lane);
#pragma unroll
      for (int ct = 0; ct < 4; ct++) {
        v2f b = load_bt(Ks, kk, ct * 16, LQ, lane);
        sf[ct] = wmma4(a, b, sf[ct]);
      }
    }
#pragma unroll
    for (int ct = 0; ct < 4; ct++)
#pragma unroll
      for (int r = 0; r < 8; r++)
        pw[(r + rofs) * LQ + ct * 16 + nn] = sf[ct][r];
    __syncthreads();

    if (lane < 16) {
      const bool band = (kbi < 3);
      float* prow = &pw[lane * LQ];
      float rmax = -1e30f;
      for (int c = 0; c < 64; c++) {
        float mm = band ? (qm * kmask[c]) : kmask[c];
        float sc = prow[c] + (1.0f - mm) * NEGV;
        prow[c] = sc;
        rmax = fmaxf(rmax, sc);
      }
      float nm = fmaxf(mrun, rmax);
      float al = __expf(mrun - nm);
      float psum = 0.f;
      for (int c = 0; c < 64; c++) {
        float p = __expf(prow[c] - nm);
        prow[c] = p;
        psum += p;
      }
      lsum = lsum * al + psum;
      mrun = nm;
      alpha_s[wave][lane] = al;
    }
    __syncthreads();

#pragma unroll
    for (int ct = 0; ct < 4; ct++)
#pragma unroll
      for (int r = 0; r < 8; r++)
        acc[ct][r] *= alpha_s[wave][r + rofs];

#pragma unroll 4
    for (int kk = 0; kk < 64; kk += 4) {
      v2f a = load_a(pw, 0, kk, LQ, lane);
#pragma unroll
      for (int ht = 0; ht < 4; ht++) {
        v2f b = load_b(Vs, kk, ht * 16, 64, lane);
        acc[ht] = wmma4(a, b, acc[ht]);
      }
    }
  }

  if (lane < 16) rl[wave][lane] = 1.0f / lsum;
  __syncthreads();

  float* obase = ctx + (((size_t)bb * H_ + hh) * S_ + qb * 64) * HD_;
#pragma unroll
  for (int ct = 0; ct < 4; ct++)
#pragma unroll
    for (int r = 0; r < 8; r++)
      obase[(wave * 16 + r + rofs) * 64 + ct * 16 + nn] =
          acc[ct][r] * rl[wave][r + rofs];
}

// ---------------- 4) output projection: h = ctx@Wo + bo + x ----------------
// ctx is in (B,H,S,HD); each 64-wide K tile maps to a single head -> contiguous rows.

__global__ __launch_bounds__(128) void outproj_kernel(const float* __restrict__ ctx,
                                                      const float* __restrict__ Wo,
                                                      const float* __restrict__ bo,
                                                      const float* __restrict__ x,
                                                      float* __restrict__ hout) {
  __shared__ float As[64 * LDA];
  __shared__ float Ws[64 * LDA];
  const int tid = threadIdx.x;
  const int wave = tid >> 5, lane = tid & 31;
  const int row0 = blockIdx.x * 64;
  const int col0 = blockIdx.y * 64;

  v8f acc[4] = {};

  for (int k0 = 0; k0 < D_; k0 += 64) {
    const int hh = k0 >> 6;
    __syncthreads();
#pragma unroll
    for (int i = 0; i < 8; i++) {
      int f = tid + i * 128;
      int r = f >> 4, c4 = f & 15;
      int row = row0 + r;
      int bb = row >> 12, ss = row & 4095;
      *(float4*)&As[r * LDA + c4 * 4] =
          *(const float4*)(ctx + (((size_t)bb * H_ + hh) * S_ + ss) * HD_ + c4 * 4);
      *(float4*)&Ws[r * LDA + c4 * 4] =
          *(const float4*)(Wo + (size_t)(k0 + r) * D_ + col0 + c4 * 4);
    }
    __syncthreads();
#pragma unroll 4
    for (int kk = 0; kk < 64; kk += 4) {
      v2f a = load_a(As, wave * 16, kk, LDA, lane);
#pragma unroll
      for (int ct = 0; ct < 4; ct++) {
        v2f b = load_b(Ws, kk, ct * 16, LDA, lane);
        acc[ct] = wmma4(a, b, acc[ct]);
      }
    }
  }

  const int nn = lane & 15;
  const int rofs = (lane >> 4) << 3;
#pragma unroll
  for (int ct = 0; ct < 4; ct++) {
    int col = col0 + ct * 16 + nn;
#pragma unroll
    for (int r = 0; r < 8; r++) {
      int row = row0 + wave * 16 + r + rofs;
      hout[(size_t)row * D_ + col] =
          acc[ct][r] + bo[col] + x[(size_t)row * D_ + col];
    }
  }
}

// ---------------- 5) LayerNorm over D=512 per row ----------------

__global__ __launch_bounds__(128) void ln_kernel(const float* __restrict__ hbuf,
                                                 const float* __restrict__ gamma,
                                                 const float* __restrict__ beta,
                                                 float* __restrict__ out) {
  __shared__ float red[128];
  const int row = blockIdx.x;
  const int tid = threadIdx.x;
  const float* hr = hbuf + (size_t)row * D_;

  float s = 0.f;
  for (int c = tid; c < D_; c += 128) s += hr[c];
  red[tid] = s;
  __syncthreads();
  for (int ofs = 64; ofs > 0; ofs >>= 1) {
    if (tid < ofs) red[tid] += red[tid + ofs];
    __syncthreads();
  }
  float mu = red[0] * (1.0f / D_);
  __syncthreads();

  float vs = 0.f;
  for (int c = tid; c < D_; c += 128) {
    float d = hr[c] - mu;
    vs += d * d;
  }
  red[tid] = vs;
  __syncthreads();
  for (int ofs = 64; ofs > 0; ofs >>= 1) {
    if (tid < ofs) red[tid] += red[tid + ofs];
    __syncthreads();
  }
  float rstd = rsqrtf(red[0] * (1.0f / D_) + 1e-12f);

  for (int c = tid; c < D_; c += 128)
    out[(size_t)row * D_ + c] = (hr[c] - mu) * rstd * gamma[c] + beta[c];
}

// ---------------- host launch ----------------

extern "C" void kernel_launch(void* const* d_in, const int* in_sizes, int n_in,
                              void* d_out, int out_size, void* d_ws, size_t ws_size,
                              hipStream_t stream) {
  const float* x    = (const float*)d_in[0];
  const float* mask = (const float*)d_in[1];
  const int*   rnd  = (const int*)d_in[2];
  const float* Wq   = (const float*)d_in[3];
  const float* Wk   = (const float*)d_in[4];
  const float* Wv   = (const float*)d_in[5];
  const float* Wo   = (const float*)d_in[6];
  const float* bo   = (const float*)d_in[7];
  const float* gam  = (const float*)d_in[8];
  const float* bet  = (const float*)d_in[9];

  float* ws = (float*)d_ws;
  const size_t SZ = (size_t)B_ * H_ * S_ * HD_;   // 4,194,304 floats
  float* qb   = ws;
  float* kb   = ws + SZ;
  float* vb   = ws + 2 * SZ;
  float* ctx  = ws + 3 * SZ;
  float* hbuf = ws + 4 * SZ;

  dim3 gg(128, 8);   // 8192/64 row tiles x 512/64 col tiles
  proj_kernel<<<gg, 128, 0, stream>>>(x, Wq, qb, 0.125f);   // HD^-0.5 = 1/8
  proj_kernel<<<gg, 128, 0, stream>>>(x, Wk, kb, 1.0f);
  proj_kernel<<<gg, 128, 0, stream>>>(x, Wv, vb, 1.0f);

  dense_attn<<<B_ * H_ * 4, 128, 0, stream>>>(qb, kb, vb, mask, ctx);
  mid_attn<<<B_ * H_ * M_, 128, 0, stream>>>(qb, kb, vb, mask, rnd, ctx);

  outproj_kernel<<<gg, 128, 0, stream>>>(ctx, Wo, bo, x, hbuf);
  ln_kernel<<<B_ * S_, 128, 0, stream>>>(hbuf, gam, bet, (float*)d_out);
}